// SparseGrid_79809082295099
// MI455X (gfx1250) — compile-verified
//
#include <hip/hip_runtime.h>

// CDNA5 / gfx1250, wave32. One wave == one query point.
// Trilinear blend out[n] = sum_c w_c * v_c[n] is routed through
// V_WMMA_F32_16X16X4_F32 as D(16x16) = A(16x4: broadcast weights) x
// B(4x16: corner values x channels) + C, chained over two K=4 groups
// (corners 0-3, 4-7) and two channel chunks (0-15, 16-27).
// Memory-bound: ~2.1 GB total traffic -> ~90us floor at 23.3 TB/s.

typedef __attribute__((ext_vector_type(2))) float v2f;
typedef __attribute__((ext_vector_type(8))) float v8f;

#define RESO      128
#define DATA_DIM  28

__global__ __launch_bounds__(256) void SparseGrid_trilerp_wmma(
    const float* __restrict__ data,    // [RESO^3][28]
    const float* __restrict__ points,  // [N][3]
    const int*   __restrict__ links,   // [RESO][RESO][RESO]
    float*       __restrict__ out,     // [N][28]
    int npoints)
{
  const int lane  = threadIdx.x & 31;
  const int waveb = threadIdx.x >> 5;                 // wave id in block
  const long long point =
      (long long)blockIdx.x * (blockDim.x >> 5) + waveb;
  if (point >= npoints) return;                       // wave-uniform: EXEC stays full

  // ---- point transform (replicated across lanes; broadcast loads) ----
  // p = pt*64 + 63.5, clip [0,127], l = min(int(p),126), wb = p-l, wa = 1-wb
  const float px = points[point * 3 + 0];
  const float py = points[point * 3 + 1];
  const float pz = points[point * 3 + 2];

  float fx = fminf(fmaxf(px * 64.0f + 63.5f, 0.0f), (float)(RESO - 1));
  float fy = fminf(fmaxf(py * 64.0f + 63.5f, 0.0f), (float)(RESO - 1));
  float fz = fminf(fmaxf(pz * 64.0f + 63.5f, 0.0f), (float)(RESO - 1));

  int lx = min((int)fx, RESO - 2);
  int ly = min((int)fy, RESO - 2);
  int lz = min((int)fz, RESO - 2);

  const float wbx = fx - (float)lx, wax = 1.0f - wbx;
  const float wby = fy - (float)ly, way = 1.0f - wby;
  const float wbz = fz - (float)lz, waz = 1.0f - wbz;

  const int hi      = lane >> 4;   // lane half: selects K offset {0,2} in A/B VGPR layout
  const int ch_lane = lane & 15;   // channel within chunk (B column / D column)

  // Corner for (group g, vgpr j, half hi): c = 4g + 2hi + j  (bit2=dx, bit1=dy, bit0=dz)
  float wv[2][2];          // [g][j] : A-matrix element this lane supplies
  float bv[2][2][2];       // [chunk][g][j] : B-matrix element this lane supplies
#pragma unroll
  for (int g = 0; g < 2; ++g) {
#pragma unroll
    for (int j = 0; j < 2; ++j) {
      const int c  = g * 4 + hi * 2 + j;
      const int dx = (c >> 2) & 1, dy = (c >> 1) & 1, dz = c & 1;
      wv[g][j] = (dx ? wbx : wax) * (dy ? wby : way) * (dz ? wbz : waz);

      const int li   = ((lx + dx) * RESO + (ly + dy)) * RESO + (lz + dz);
      const int link = links[li];                      // gather (broadcast within half)
#pragma unroll
      for (int t = 0; t < 2; ++t) {
        const int ch = t * 16 + ch_lane;
        // lanes 0-15 of VGPR j read 64B contiguous of corner row -> coalesced
        bv[t][g][j] = (link >= 0 && ch < DATA_DIM)
                          ? data[(size_t)link * DATA_DIM + ch]
                          : 0.0f;
      }
    }
  }

  // ---- WMMA: two chained 16x16x4 f32 per channel chunk ----
  float res[2];
#pragma unroll
  for (int t = 0; t < 2; ++t) {
    v2f a0; a0[0] = wv[0][0];    a0[1] = wv[0][1];     // A: K = {0,1}+2*hi -> corners 0-3
    v2f b0; b0[0] = bv[t][0][0]; b0[1] = bv[t][0][1];  // B: rows = corners 0-3
    v8f acc = {};
    acc = __builtin_amdgcn_wmma_f32_16x16x4_f32(
        false, a0, false, b0, (short)0, acc, false, false);
    v2f a1; a1[0] = wv[1][0];    a1[1] = wv[1][1];     // corners 4-7
    v2f b1; b1[0] = bv[t][1][0]; b1[1] = bv[t][1][1];
    acc = __builtin_amdgcn_wmma_f32_16x16x4_f32(
        false, a1, false, b1, (short)0, acc, false, false);
    // Every row of D equals out[chunk]; VGPR0 lanes 0-15 = D[M=0][N=lane]
    res[t] = acc[0];
  }

  // ---- store (divergent OK: after all WMMAs) ----
  if (lane < 16) {
    float* o = out + point * DATA_DIM;
    o[ch_lane] = res[0];                // channels 0-15
    const int ch1 = 16 + ch_lane;
    if (ch1 < DATA_DIM) o[ch1] = res[1];  // channels 16-27
  }
}

extern "C" void kernel_launch(void* const* d_in, const int* in_sizes, int n_in,
                              void* d_out, int out_size, void* d_ws, size_t ws_size,
                              hipStream_t stream) {
  const float* data   = (const float*)d_in[0];
  const float* points = (const float*)d_in[1];
  const int*   links  = (const int*)d_in[2];
  float*       out    = (float*)d_out;

  const int npoints = in_sizes[1] / 3;          // points is [N][3]
  const int waves_per_block = 256 / 32;         // 8 points per block
  const int grid = (npoints + waves_per_block - 1) / waves_per_block;

  SparseGrid_trilerp_wmma<<<grid, 256, 0, stream>>>(data, points, links, out,
                                                    npoints);
}